// SinStack_52381421142257
// MI455X (gfx1250) — compile-verified
//
#include <hip/hip_runtime.h>
#include <math.h>

// SinStack additive synthesis + Hann overlap-add for gfx1250 (MI455X).
//
// Chirp factorization: w = w0 + 16*w1 turns the per-frame harmonic sum into a
// 16 x (2N) x 16 GEMM executed with V_WMMA_F32_16X16X4_F32:
//   A[w0, 2n+0] =  a*cos(phi + omega*(w0-512))   (Re u)
//   A[w0, 2n+1] =  a*sin(phi + omega*(w0-512))   (Im u)
//   B[2n+0, w1] =  cos(16*omega*w1)              (Re v)
//   B[2n+1, w1] = -sin(16*omega*w1)              (-Im v)
//   D = A*B  =>  sum_n a*cos(phi + omega*(w - 512))
// Each wave owns one 512-sample output segment [t*512,(t+1)*512): second half
// of frame t + first half of frame t+1; Hann applied in registers; one store
// per output sample (no atomics, no workspace).

typedef float v2f __attribute__((ext_vector_type(2)));
typedef float v8f __attribute__((ext_vector_type(8)));

#define HOPC    512
#define WINC    1024
#define NHAR    100
#define TFR     400
#define NBATCH  4

__global__ __launch_bounds__(256) void sinstack_wmma_kernel(
    const float* __restrict__ ampl,
    const float* __restrict__ phase,
    const float* __restrict__ f0,
    float* __restrict__ out)
{
    const int lane      = threadIdx.x & 31;
    const int waveInBlk = threadIdx.x >> 5;
    const int wid       = blockIdx.x * (blockDim.x >> 5) + waveInBlk; // [0,1600)
    const int b         = wid / TFR;
    const int t         = wid - b * TFR;
    const bool hasNext  = (t + 1) < TFR;

    const int half = lane >> 4;   // selects which of the 2 harmonics in a K=4 chunk
    const int col  = lane & 15;   // A: row w0 ; B: column w1 ; C/D: column

    const float OMEGA_SCALE = 6.283185307179586f / 44100.0f; // 2*pi/SR

    v8f acc0 = {0.f,0.f,0.f,0.f,0.f,0.f,0.f,0.f}; // frame t+1, w1 base 0   (w in [  0,256))
    v8f acc1 = {0.f,0.f,0.f,0.f,0.f,0.f,0.f,0.f}; // frame t+1, w1 base 16  (w in [256,512))
    v8f acc2 = {0.f,0.f,0.f,0.f,0.f,0.f,0.f,0.f}; // frame t,   w1 base 32  (w in [512,768))
    v8f acc3 = {0.f,0.f,0.f,0.f,0.f,0.f,0.f,0.f}; // frame t,   w1 base 48  (w in [768,1024))

    const int baseT = (b * NHAR) * TFR + t;   // + n*TFR selects harmonic n

    for (int chunk = 0; chunk < NHAR / 2; ++chunk) {
        const int n    = 2 * chunk + half;        // this lane-half's harmonic
        const int idxT = baseT + n * TFR;

        // ----- frame t : output tiles for w in [512, 1024) -----
        {
            const float a  = ampl[idxT];
            const float ph = phase[idxT];
            const float om = f0[idxT] * OMEGA_SCALE;

            // A = a * e^{i(ph + om*(w0-512))},   w0 = col
            const float thA = ph + om * (float)(col - HOPC);
            v2f A = { a * __cosf(thA), a * __sinf(thA) };

            // B seed at w1 base 32: v = e^{i*om*16*(32+col)}
            const float thB = om * (float)(512 + 16 * col);
            float c = __cosf(thB);
            float s = __sinf(thB);
            v2f Bm = { c, -s };
            acc2 = __builtin_amdgcn_wmma_f32_16x16x4_f32(
                       false, A, false, Bm, (short)0, acc2, false, false);

            // rotate by e^{i*256*om} -> w1 base 48
            const float thR = om * 256.0f;
            const float cr = __cosf(thR);
            const float sr = __sinf(thR);
            const float c2 = c * cr - s * sr;
            const float s2 = s * cr + c * sr;
            Bm.x = c2; Bm.y = -s2;
            acc3 = __builtin_amdgcn_wmma_f32_16x16x4_f32(
                       false, A, false, Bm, (short)0, acc3, false, false);
        }

        // ----- frame t+1 : output tiles for w in [0, 512) -----
        if (hasNext) { // wave-uniform branch: EXEC stays all-ones for WMMA
            const int idxT1 = idxT + 1;
            const float a  = ampl[idxT1];
            const float ph = phase[idxT1];
            const float om = f0[idxT1] * OMEGA_SCALE;

            const float thA = ph + om * (float)(col - HOPC);
            v2f A = { a * __cosf(thA), a * __sinf(thA) };

            // B seed at w1 base 0
            const float thB = om * (float)(16 * col);
            float c = __cosf(thB);
            float s = __sinf(thB);
            v2f Bm = { c, -s };
            acc0 = __builtin_amdgcn_wmma_f32_16x16x4_f32(
                       false, A, false, Bm, (short)0, acc0, false, false);

            const float thR = om * 256.0f;
            const float cr = __cosf(thR);
            const float sr = __sinf(thR);
            const float c2 = c * cr - s * sr;
            const float s2 = s * cr + c * sr;
            Bm.x = c2; Bm.y = -s2;
            acc1 = __builtin_amdgcn_wmma_f32_16x16x4_f32(
                       false, A, false, Bm, (short)0, acc1, false, false);
        }
    }

    // ----- Hann window + overlap-add, one store per output sample -----
    // C/D layout: element j -> M = j + 8*half (= w0), N = col (= w1 offset)
    // frame t   tile (w1 base 32/48): w = 512 + r, r = m + 16*col (+256)
    // frame t+1 tile (w1 base  0/16): w = r          (same r, same lane/elem)
    // hann[512+r] = 0.5 + 0.5*cos(pi*r/512) ; hann[r] = 0.5 - 0.5*cos(pi*r/512)
    const int outBase = b * (TFR * HOPC) + t * HOPC;
    const float PI_OVER_HOP = 3.14159265358979f / 512.0f;

#pragma unroll
    for (int j = 0; j < 8; ++j) {
        const int m  = j + 8 * half;
        const int r1 = m + 16 * col;            // [0,256)
        const float cw1 = __cosf((float)r1 * PI_OVER_HOP);
        out[outBase + r1] = acc2[j] * (0.5f + 0.5f * cw1)
                          + acc0[j] * (0.5f - 0.5f * cw1);

        const int r2 = r1 + 256;                // [256,512)
        const float cw2 = __cosf((float)r2 * PI_OVER_HOP);
        out[outBase + r2] = acc3[j] * (0.5f + 0.5f * cw2)
                          + acc1[j] * (0.5f - 0.5f * cw2);
    }
}

extern "C" void kernel_launch(void* const* d_in, const int* in_sizes, int n_in,
                              void* d_out, int out_size, void* d_ws, size_t ws_size,
                              hipStream_t stream) {
    (void)in_sizes; (void)n_in; (void)d_ws; (void)ws_size; (void)out_size;
    const float* ampl  = (const float*)d_in[0];
    const float* phase = (const float*)d_in[1];
    const float* f0    = (const float*)d_in[2];
    float* out = (float*)d_out;

    // 1600 (b,t) segments, one wave32 each -> 200 blocks x 256 threads (8 waves)
    dim3 grid(200), block(256);
    hipLaunchKernelGGL(sinstack_wmma_kernel, grid, block, 0, stream,
                       ampl, phase, f0, out);
}